// SpixFeatureProjection_18949395710734
// MI455X (gfx1250) — compile-verified
//
#include <hip/hip_runtime.h>
#include <hip/hip_bf16.h>

typedef __attribute__((ext_vector_type(2))) float v2f;
typedef __attribute__((ext_vector_type(8))) float v8f;

#define Bn    16
#define Fn    32
#define Nn    65536          // 256*256 pixels
#define OUTD  8
#define TPB   256
#define LP    260            // padded pixel stride in LDS (dwords) -> conflict-free frags
#define CHUNKS 16            // pass-1 blocks along pixel dim
#define PIX_PER_BLK (Nn / CHUNKS)   // 4096
#define ITERS (PIX_PER_BLK / TPB)   // 16

// ---------------------------------------------------------------------------
// Pass 0: zero the per-batch Gram accumulators in workspace
// ---------------------------------------------------------------------------
__global__ void k_zero(float* __restrict__ g, int n) {
    int i = blockIdx.x * blockDim.x + threadIdx.x;
    if (i < n) g[i] = 0.0f;
}

// ---------------------------------------------------------------------------
// Pass 1: fused per-pixel normalization + Gram (G = Xn^T Xn) via fp32 WMMA.
// LDS layout: lds[f*LP + p]  (f=0..31 features, p=0..255 pixels of the tile).
// A frag (16x4, M=feat,K=pix):  lane L -> m=L&15, K = k0+2*(L>>4)+{0,1}
// B frag (4x16, K=pix,N=feat):  lane L -> n=L&15, K = k0+(L>>4)+{0,2}
// G is symmetric: accumulate tiles (0,0),(0,1),(1,1) only.
// ---------------------------------------------------------------------------
__global__ __launch_bounds__(TPB) void k_gram(const float* __restrict__ x,
                                              float* __restrict__ G) {
    __shared__ float lds[Fn * LP];   // 8320 floats = 33280 B (also reused for reduction)
    const int tid  = threadIdx.x;
    const int lane = tid & 31;
    const int wave = tid >> 5;
    const int b    = blockIdx.y;
    const float* xb = x + (size_t)b * Fn * Nn;

    v8f acc00 = {}; v8f acc01 = {}; v8f acc11 = {};

    for (int it = 0; it < ITERS; ++it) {
        const int p = blockIdx.x * PIX_PER_BLK + it * TPB + tid;

        // coalesced load of this pixel's 32 features (imm offsets f*Nn*4 fit i24)
        float xv[Fn];
        float s = 0.0f, ss = 0.0f;
        #pragma unroll
        for (int f = 0; f < Fn; ++f) {
            float v = xb[(size_t)f * Nn + p];
            xv[f] = v; s += v; ss += v * v;
        }
        const float mean = s * (1.0f / Fn);
        const float rs   = rsqrtf((ss - s * mean) * (1.0f / (Fn - 1)));
        #pragma unroll
        for (int f = 0; f < Fn; ++f)
            lds[f * LP + tid] = (xv[f] - mean) * rs;
        __syncthreads();

        // each wave consumes its own 32 pixels of the tile: 8 K-steps of 4
        const int m  = lane & 15;
        const int g2 = lane >> 4;
        #pragma unroll
        for (int s4 = 0; s4 < 8; ++s4) {
            const int k0 = wave * 32 + s4 * 4;
            const int ka = k0 + 2 * g2;
            const int kb = k0 + g2;
            v2f alo, ahi, blo, bhi;
            alo.x = lds[m * LP + ka];            alo.y = lds[m * LP + ka + 1];
            ahi.x = lds[(m + 16) * LP + ka];     ahi.y = lds[(m + 16) * LP + ka + 1];
            blo.x = lds[m * LP + kb];            blo.y = lds[m * LP + kb + 2];
            bhi.x = lds[(m + 16) * LP + kb];     bhi.y = lds[(m + 16) * LP + kb + 2];
            acc00 = __builtin_amdgcn_wmma_f32_16x16x4_f32(false, alo, false, blo,
                                                          (short)0, acc00, false, false);
            acc01 = __builtin_amdgcn_wmma_f32_16x16x4_f32(false, alo, false, bhi,
                                                          (short)0, acc01, false, false);
            acc11 = __builtin_amdgcn_wmma_f32_16x16x4_f32(false, ahi, false, bhi,
                                                          (short)0, acc11, false, false);
        }
        __syncthreads();
    }

    // cross-wave reduction through LDS, then atomic add into the global Gram.
    // C/D layout: VGPR v, lanes 0-15 -> (M=v, N=lane); lanes 16-31 -> (M=v+8, N=lane-16)
    {
        const int idxBase = (8 * (lane >> 4)) * 16 + (lane & 15);
        const int base = wave * 768;
        #pragma unroll
        for (int v = 0; v < 8; ++v) {
            const int idx = idxBase + v * 16;
            lds[base + 0 * 256 + idx] = acc00[v];
            lds[base + 1 * 256 + idx] = acc01[v];
            lds[base + 2 * 256 + idx] = acc11[v];
        }
    }
    __syncthreads();

    float* Gb = G + b * (Fn * Fn);
    #pragma unroll
    for (int t = 0; t < 3; ++t) {
        float sum = 0.0f;
        #pragma unroll
        for (int w = 0; w < 8; ++w) sum += lds[w * 768 + t * 256 + tid];
        const int r = tid >> 4, c = tid & 15;
        if (t == 0) {
            atomicAdd(&Gb[r * Fn + c], sum);
        } else if (t == 1) {                       // off-diagonal block + its mirror
            atomicAdd(&Gb[r * Fn + 16 + c], sum);
            atomicAdd(&Gb[(16 + c) * Fn + r], sum);
        } else {
            atomicAdd(&Gb[(16 + r) * Fn + 16 + c], sum);
        }
    }
}

// ---------------------------------------------------------------------------
// Pass 2: one wave per batch, cyclic Jacobi eigensolve of the 32x32 Gram.
// Single-wave workgroup: barriers lower to S_NOP per ISA, lanes are lockstep.
// Writes top-8 eigenvectors (desc eigenvalue order) as V8[b][f][j].
// ---------------------------------------------------------------------------
__global__ __launch_bounds__(32) void k_eig(const float* __restrict__ G,
                                            float* __restrict__ V8) {
    __shared__ float A[Fn][Fn + 1];
    __shared__ float V[Fn][Fn + 1];
    __shared__ float ev[Fn];
    __shared__ int   idxr[OUTD];
    const int l = threadIdx.x;
    const int b = blockIdx.x;
    const float* Gb = G + b * Fn * Fn;

    for (int r = 0; r < Fn; ++r) {
        A[r][l] = Gb[r * Fn + l];
        V[r][l] = (r == l) ? 1.0f : 0.0f;
    }
    __syncthreads();

    for (int sweep = 0; sweep < 12; ++sweep) {
        for (int p = 0; p < Fn - 1; ++p) {
            for (int q = p + 1; q < Fn; ++q) {
                const float apq = A[p][q];
                const float app = A[p][p];
                const float aqq = A[q][q];
                if (fabsf(apq) > 1e-20f) {           // uniform branch (LDS-sourced)
                    const float tau = (aqq - app) / (2.0f * apq);
                    const float t   = copysignf(1.0f, tau) /
                                      (fabsf(tau) + sqrtf(1.0f + tau * tau));
                    const float c   = rsqrtf(1.0f + t * t);
                    const float s   = t * c;
                    const float rp = A[p][l], rq = A[q][l];
                    __syncthreads();
                    A[p][l] = c * rp - s * rq;
                    A[q][l] = s * rp + c * rq;
                    __syncthreads();
                    const float cp = A[l][p], cq = A[l][q];
                    __syncthreads();
                    A[l][p] = c * cp - s * cq;
                    A[l][q] = s * cp + c * cq;
                    const float vp = V[l][p], vq = V[l][q];
                    V[l][p] = c * vp - s * vq;
                    V[l][q] = s * vp + c * vq;
                }
                __syncthreads();
            }
        }
    }

    ev[l] = A[l][l];
    __syncthreads();
    const float lam = ev[l];
    int rank = 0;
    for (int j = 0; j < Fn; ++j) {
        const float lj = ev[j];
        rank += ((lj > lam) || (lj == lam && j < l)) ? 1 : 0;
    }
    if (rank < OUTD) idxr[rank] = l;
    __syncthreads();

    float* out = V8 + b * (Fn * OUTD);
    #pragma unroll
    for (int r = 0; r < OUTD; ++r)
        out[l * OUTD + r] = V[l][idxr[r]];
}

// ---------------------------------------------------------------------------
// Pass 3: out[b][j][p] = sum_f Xn[b][p][f] * V8[b][f][j]
// V8 indices are uniform -> scalar loads feeding v_fmac; stores coalesced.
// x is L2-resident (134 MB < 192 MB L2) after pass 1.
// ---------------------------------------------------------------------------
__global__ __launch_bounds__(TPB) void k_proj(const float* __restrict__ x,
                                              const float* __restrict__ V8,
                                              float* __restrict__ out) {
    const int b = blockIdx.y;
    const int p = blockIdx.x * TPB + threadIdx.x;
    const float* xb = x + (size_t)b * Fn * Nn;
    const float* vb = V8 + b * Fn * OUTD;

    float xv[Fn];
    float s = 0.0f, ss = 0.0f;
    #pragma unroll
    for (int f = 0; f < Fn; ++f) {
        float v = xb[(size_t)f * Nn + p];
        xv[f] = v; s += v; ss += v * v;
    }
    const float mean = s * (1.0f / Fn);
    const float rs   = rsqrtf((ss - s * mean) * (1.0f / (Fn - 1)));

    float acc[OUTD] = {};
    #pragma unroll
    for (int f = 0; f < Fn; ++f) {
        const float v = (xv[f] - mean) * rs;
        #pragma unroll
        for (int j = 0; j < OUTD; ++j)
            acc[j] = fmaf(v, vb[f * OUTD + j], acc[j]);
    }

    float* ob = out + (size_t)b * OUTD * Nn;
    #pragma unroll
    for (int j = 0; j < OUTD; ++j)
        ob[(size_t)j * Nn + p] = acc[j];
}

// ---------------------------------------------------------------------------
extern "C" void kernel_launch(void* const* d_in, const int* in_sizes, int n_in,
                              void* d_out, int out_size, void* d_ws, size_t ws_size,
                              hipStream_t stream) {
    const float* x  = (const float*)d_in[0];
    float* out      = (float*)d_out;
    float* G        = (float*)d_ws;                    // 16*32*32 floats = 64 KB
    float* V8       = (float*)d_ws + Bn * Fn * Fn;     // 16*32*8 floats  = 16 KB

    k_zero<<<dim3((Bn * Fn * Fn + 255) / 256), dim3(256), 0, stream>>>(G, Bn * Fn * Fn);
    k_gram<<<dim3(CHUNKS, Bn), dim3(TPB), 0, stream>>>(x, G);
    k_eig <<<dim3(Bn), dim3(32), 0, stream>>>(G, V8);
    k_proj<<<dim3(Nn / TPB, Bn), dim3(TPB), 0, stream>>>(x, V8, out);
}